// RelationConv_90185723281819
// MI455X (gfx1250) — compile-verified
//
#include <hip/hip_runtime.h>
#include <math.h>

#define BATCH 4
#define NPTS  4096
#define CIN   61
#define FIN   64    // CIN + 3
#define FOUT  128
#define KNN   16
#define WPB   8     // waves per block in mlp_pool

typedef float v2f __attribute__((ext_vector_type(2)));
typedef float v8f __attribute__((ext_vector_type(8)));

// D = A(16x4 f32) * B(4x16 f32) + C(16x16 f32)   -- V_WMMA_F32_16X16X4_F32
__device__ __forceinline__ v8f wmma4(float a0, float a1, float b0, float b1, v8f c) {
  v2f A = {a0, a1};
  v2f B = {b0, b1};
  return __builtin_amdgcn_wmma_f32_16x16x4_f32(false, A, false, B, (short)0, c,
                                               false, false);
}

// ---------------------------------------------------------------------------
// Kernel 1: brute-force KNN (K=16 smallest squared distances, ties -> lower idx)
// One block handles 256 queries of one batch; entire pos[b] staged in LDS (48KB).
// ---------------------------------------------------------------------------
__global__ void knn_kernel(const float* __restrict__ pos, int* __restrict__ idx) {
  __shared__ float sp[NPTS * 3];
  const int b = blockIdx.y;
  const float* pb = pos + (size_t)b * NPTS * 3;
  for (int i = threadIdx.x; i < NPTS * 3; i += blockDim.x) sp[i] = pb[i];
  __syncthreads();

  const int q = blockIdx.x * blockDim.x + threadIdx.x;
  const float qx = sp[q * 3 + 0], qy = sp[q * 3 + 1], qz = sp[q * 3 + 2];

  float bd[KNN];
  int   bi[KNN];
#pragma unroll
  for (int j = 0; j < KNN; ++j) { bd[j] = 3.4e38f; bi[j] = 0; }

  for (int m = 0; m < NPTS; ++m) {
    const float dx = sp[m * 3 + 0] - qx;
    const float dy = sp[m * 3 + 1] - qy;
    const float dz = sp[m * 3 + 2] - qz;
    const float d = dx * dx + dy * dy + dz * dz;
    if (d < bd[KNN - 1]) {                 // strict < keeps earlier index on tie
      bd[KNN - 1] = d; bi[KNN - 1] = m;
#pragma unroll
      for (int j = KNN - 1; j > 0; --j) {  // bubble new element into place
        if (bd[j] < bd[j - 1]) {
          float td = bd[j]; bd[j] = bd[j - 1]; bd[j - 1] = td;
          int   ti = bi[j]; bi[j] = bi[j - 1]; bi[j - 1] = ti;
        }
      }
    }
  }
  int* op = idx + ((size_t)b * NPTS + q) * KNN;
#pragma unroll
  for (int j = 0; j < KNN; ++j) op[j] = bi[j];
}

// ---------------------------------------------------------------------------
// Kernel 2: per-query relation MLP (10->32->64->64) via f32 WMMA, then
// weights*feat, max over the 16 neighbors, relu  ->  pooled[B*N][64].
// One wave32 per query: its 16 neighbors form exactly one M=16 WMMA tile.
// ---------------------------------------------------------------------------
__global__ void mlp_pool_kernel(
    const float* __restrict__ x, const float* __restrict__ pos,
    const int* __restrict__ idx,
    const float* __restrict__ rw1, const float* __restrict__ rb1,
    const float* __restrict__ rw2, const float* __restrict__ rb2,
    const float* __restrict__ rw3, const float* __restrict__ rb3,
    float* __restrict__ pooled) {
  __shared__ float s_rel[WPB][16 * 12];   // relation, K padded 10 -> 12
  __shared__ float s_h1[WPB][16 * 32];
  __shared__ float s_h2[WPB][16 * 64];
  __shared__ int   s_idx[WPB][16];

  const int tid   = threadIdx.x;
  const int w     = tid >> 5;
  const int lane  = tid & 31;
  const int row16 = lane & 15;
  const int hi    = lane >> 4;     // lane half
  const int koff  = hi * 2;        // f32 WMMA A/B: v0 -> K 0/2, v1 -> K 1/3
  const int roff  = hi * 8;        // f32 WMMA D: vgpr j -> row j (+8 upper half)

  const int q = blockIdx.x * WPB + w;      // 0 .. B*N-1 (exact)
  const int b = q >> 12;
  const int n = q & (NPTS - 1);

  const float* pb = pos + (size_t)b * NPTS * 3;
  const float cx = pb[n * 3 + 0], cy = pb[n * 3 + 1], cz = pb[n * 3 + 2];

  // ---- gather neighbors, build relation rows ----
  if (lane < 16) {
    const int j = idx[(size_t)q * KNN + lane];
    s_idx[w][lane] = j;
    const float gx_ = pb[j * 3 + 0], gy = pb[j * 3 + 1], gz = pb[j * 3 + 2];
    const float rx = gx_ - cx, ry = gy - cy, rz = gz - cz;
    const float s = rx * rx + ry * ry + rz * rz;
    const float dis = (s > 0.f) ? sqrtf(s) : 0.f;
    float* rr = &s_rel[w][lane * 12];
    rr[0] = cx;  rr[1] = cy;  rr[2] = cz;
    rr[3] = gx_; rr[4] = gy;  rr[5] = gz;
    rr[6] = rx;  rr[7] = ry;  rr[8] = rz;
    rr[9] = dis; rr[10] = 0.f; rr[11] = 0.f;
  }
  __syncthreads();

  // ---- layer 1: rel(16x10) @ rw1(10x32) + rb1, relu -> h1(16x32) ----
  for (int nt = 0; nt < 2; ++nt) {
    const int col = nt * 16 + row16;
    v8f acc = {0, 0, 0, 0, 0, 0, 0, 0};
#pragma unroll
    for (int c = 0; c < 3; ++c) {
      const int k0 = c * 4 + koff;
      const float a0 = s_rel[w][row16 * 12 + k0];
      const float a1 = s_rel[w][row16 * 12 + k0 + 1];
      const float b0 = (k0 < 10)     ? rw1[k0 * 32 + col]       : 0.f;
      const float b1 = (k0 + 1 < 10) ? rw1[(k0 + 1) * 32 + col] : 0.f;
      acc = wmma4(a0, a1, b0, b1, acc);
    }
    const float bias = rb1[col];
#pragma unroll
    for (int j = 0; j < 8; ++j) {
      const float v = acc[j] + bias;
      s_h1[w][(j + roff) * 32 + col] = v > 0.f ? v : 0.f;
    }
  }
  __syncthreads();

  // ---- layer 2: h1(16x32) @ rw2(32x64) + rb2, relu -> h2(16x64) ----
  for (int nt = 0; nt < 4; ++nt) {
    const int col = nt * 16 + row16;
    v8f acc = {0, 0, 0, 0, 0, 0, 0, 0};
#pragma unroll
    for (int c = 0; c < 8; ++c) {
      const int k0 = c * 4 + koff;
      const float a0 = s_h1[w][row16 * 32 + k0];
      const float a1 = s_h1[w][row16 * 32 + k0 + 1];
      const float b0 = rw2[k0 * 64 + col];
      const float b1 = rw2[(k0 + 1) * 64 + col];
      acc = wmma4(a0, a1, b0, b1, acc);
    }
    const float bias = rb2[col];
#pragma unroll
    for (int j = 0; j < 8; ++j) {
      const float v = acc[j] + bias;
      s_h2[w][(j + roff) * 64 + col] = v > 0.f ? v : 0.f;
    }
  }
  __syncthreads();

  // ---- layer 3 + pooling: W = h2 @ rw3 + rb3; max_k(W * feat); relu ----
  const float* xb = x + (size_t)b * NPTS * CIN;
  for (int nt = 0; nt < 4; ++nt) {
    const int col = nt * 16 + row16;
    v8f acc = {0, 0, 0, 0, 0, 0, 0, 0};
#pragma unroll
    for (int c = 0; c < 16; ++c) {
      const int k0 = c * 4 + koff;
      const float a0 = s_h2[w][row16 * 64 + k0];
      const float a1 = s_h2[w][row16 * 64 + k0 + 1];
      const float b0 = rw3[k0 * 64 + col];
      const float b1 = rw3[(k0 + 1) * 64 + col];
      acc = wmma4(a0, a1, b0, b1, acc);
    }
    const float bias = rb3[col];
    float m = -3.4e38f;
#pragma unroll
    for (int j = 0; j < 8; ++j) {
      const int row = j + roff;
      const int jdx = s_idx[w][row];
      const float f = (col < CIN) ? xb[(size_t)jdx * CIN + col]
                                  : pb[jdx * 3 + (col - CIN)];
      const float v = (acc[j] + bias) * f;
      m = v > m ? v : m;
    }
    // combine the two lane-halves (rows 0-7 with rows 8-15 of same column)
    const float other = __shfl_xor(m, 16, 32);
    m = other > m ? other : m;
    m = m > 0.f ? m : 0.f;
    if (hi == 0) pooled[(size_t)q * FIN + col] = m;
  }
}

// ---------------------------------------------------------------------------
// Kernel 3: feat1 = pooled(16384x64) @ fw1(64x128) + fb1   (f32 WMMA)
// Block = 8 waves; one M-tile of 16 rows; wave w owns output cols [16w,16w+16).
// ---------------------------------------------------------------------------
__global__ void gemm1_kernel(const float* __restrict__ A,
                             const float* __restrict__ Wg,
                             const float* __restrict__ bias,
                             float* __restrict__ out) {
  __shared__ float As[16 * FIN];
  const int tid = threadIdx.x;
  const int w = tid >> 5, lane = tid & 31;
  const int row16 = lane & 15, hi = lane >> 4, koff = hi * 2, roff = hi * 8;
  const size_t m0 = (size_t)blockIdx.x * 16;

  for (int i = tid; i < 16 * FIN; i += blockDim.x) As[i] = A[m0 * FIN + i];
  __syncthreads();

  const int col = w * 16 + row16;
  v8f acc = {0, 0, 0, 0, 0, 0, 0, 0};
#pragma unroll
  for (int c = 0; c < 16; ++c) {
    const int k0 = c * 4 + koff;
    acc = wmma4(As[row16 * FIN + k0], As[row16 * FIN + k0 + 1],
                Wg[k0 * FOUT + col], Wg[(k0 + 1) * FOUT + col], acc);
  }
  const float bb = bias[col];
#pragma unroll
  for (int j = 0; j < 8; ++j)
    out[(m0 + j + roff) * FOUT + col] = acc[j] + bb;
}

// ---------------------------------------------------------------------------
// Kernel 4: per-channel BN stats -> affine (scale, shift) s.t. y = x*sc + sh
// ---------------------------------------------------------------------------
__global__ void bnstats_kernel(const float* __restrict__ src, int nrows,
                               const float* __restrict__ gamma,
                               const float* __restrict__ beta,
                               float* __restrict__ ss, int nch) {
  __shared__ float s_sum[256];
  __shared__ float s_sq[256];
  const int c = blockIdx.x;
  const int t = threadIdx.x;
  float s = 0.f, q2 = 0.f;
  for (int r = t; r < nrows; r += blockDim.x) {
    const float v = src[(size_t)r * nch + c];
    s += v; q2 += v * v;
  }
  s_sum[t] = s; s_sq[t] = q2;
  __syncthreads();
  for (int o = 128; o > 0; o >>= 1) {
    if (t < o) { s_sum[t] += s_sum[t + o]; s_sq[t] += s_sq[t + o]; }
    __syncthreads();
  }
  if (t == 0) {
    const float mean = s_sum[0] / (float)nrows;
    const float var  = s_sq[0] / (float)nrows - mean * mean;
    const float rstd = rsqrtf(var + 1e-5f);
    const float sc = rstd * gamma[c];
    ss[c * 2]     = sc;
    ss[c * 2 + 1] = beta[c] - mean * sc;
  }
}

// ---------------------------------------------------------------------------
// Kernel 5: feat2 = relu(bn1(feat1)) @ fw2(128x128) + fb2   (f32 WMMA)
// BN1 + relu applied while staging the A tile.
// ---------------------------------------------------------------------------
__global__ void gemm2_kernel(const float* __restrict__ F1,
                             const float* __restrict__ ss,
                             const float* __restrict__ Wg,
                             const float* __restrict__ bias,
                             float* __restrict__ out) {
  __shared__ float As[16 * FOUT];
  const int tid = threadIdx.x;
  const int w = tid >> 5, lane = tid & 31;
  const int row16 = lane & 15, hi = lane >> 4, koff = hi * 2, roff = hi * 8;
  const size_t m0 = (size_t)blockIdx.x * 16;

  for (int i = tid; i < 16 * FOUT; i += blockDim.x) {
    const int c = i & (FOUT - 1);
    const float v = F1[m0 * FOUT + i] * ss[c * 2] + ss[c * 2 + 1];
    As[i] = v > 0.f ? v : 0.f;
  }
  __syncthreads();

  const int col = w * 16 + row16;
  v8f acc = {0, 0, 0, 0, 0, 0, 0, 0};
#pragma unroll
  for (int c = 0; c < 32; ++c) {
    const int k0 = c * 4 + koff;
    acc = wmma4(As[row16 * FOUT + k0], As[row16 * FOUT + k0 + 1],
                Wg[k0 * FOUT + col], Wg[(k0 + 1) * FOUT + col], acc);
  }
  const float bb = bias[col];
#pragma unroll
  for (int j = 0; j < 8; ++j)
    out[(m0 + j + roff) * FOUT + col] = acc[j] + bb;
}

// ---------------------------------------------------------------------------
// Kernel 6: final BN2 (no relu) -> d_out
// ---------------------------------------------------------------------------
__global__ void bnapply_kernel(const float* __restrict__ src,
                               const float* __restrict__ ss,
                               float* __restrict__ out, int total, int nch) {
  const int i = blockIdx.x * blockDim.x + threadIdx.x;
  if (i < total) {
    const int c = i & (nch - 1);
    out[i] = src[i] * ss[c * 2] + ss[c * 2 + 1];
  }
}

// ---------------------------------------------------------------------------
extern "C" void kernel_launch(void* const* d_in, const int* in_sizes, int n_in,
                              void* d_out, int out_size, void* d_ws,
                              size_t ws_size, hipStream_t stream) {
  const float* x   = (const float*)d_in[0];
  const float* pos = (const float*)d_in[1];
  const float* rw1 = (const float*)d_in[2];
  const float* rb1 = (const float*)d_in[3];
  const float* rw2 = (const float*)d_in[4];
  const float* rb2 = (const float*)d_in[5];
  const float* rw3 = (const float*)d_in[6];
  const float* rb3 = (const float*)d_in[7];
  const float* fw1 = (const float*)d_in[8];
  const float* fb1 = (const float*)d_in[9];
  const float* g1  = (const float*)d_in[10];
  const float* b1  = (const float*)d_in[11];
  const float* fw2 = (const float*)d_in[12];
  const float* fb2 = (const float*)d_in[13];
  const float* g2  = (const float*)d_in[14];
  const float* b2  = (const float*)d_in[15];
  float* out = (float*)d_out;

  const int M = BATCH * NPTS;  // 16384

  char* ws = (char*)d_ws;
  int*   idx    = (int*)ws;                                   // M*16 ints
  float* pooled = (float*)(ws + (size_t)M * KNN * sizeof(int));
  float* feat1  = pooled + (size_t)M * FIN;
  float* feat2  = feat1 + (size_t)M * FOUT;
  float* ss1    = feat2 + (size_t)M * FOUT;
  float* ss2    = ss1 + FOUT * 2;

  knn_kernel<<<dim3(NPTS / 256, BATCH), 256, 0, stream>>>(pos, idx);
  mlp_pool_kernel<<<M / WPB, 32 * WPB, 0, stream>>>(x, pos, idx, rw1, rb1, rw2,
                                                    rb2, rw3, rb3, pooled);
  gemm1_kernel<<<M / 16, 256, 0, stream>>>(pooled, fw1, fb1, feat1);
  bnstats_kernel<<<FOUT, 256, 0, stream>>>(feat1, M, g1, b1, ss1, FOUT);
  gemm2_kernel<<<M / 16, 256, 0, stream>>>(feat1, ss1, fw2, fb2, feat2);
  bnstats_kernel<<<FOUT, 256, 0, stream>>>(feat2, M, g2, b2, ss2, FOUT);
  bnapply_kernel<<<(M * FOUT + 255) / 256, 256, 0, stream>>>(feat2, ss2, out,
                                                             M * FOUT, FOUT);
}